// MaskRCNN_16913581211763
// MI455X (gfx1250) — compile-verified
//
#include <hip/hip_runtime.h>

// 3D ROI-Align (Mask R-CNN): 512 boxes, 256 channels, 7x7x7 trilinear pooled
// output, two pyramid levels (64^3 / 32^3) selected per box by roi_level.
// Memory-bound gather kernel (~0.4 flop/byte) -> matrix ops are the wrong
// tool; optimize the gather/store path and use gfx1250 global_prefetch_b8.

#define P      7
#define NPOS   343   // 7*7*7
#define BLOCK  352   // 11 wave32s >= NPOS
#define CPB    32    // channels per block (compile-time for full unroll/fold)

__device__ __forceinline__ void axis_interp(float lo, float hi, int i, float dm1,
                                            int& i0, int& i1, float& w) {
  // Match reference: c = lo + (t * (L - 1)) / 6, L = hi - lo  (f32 op order)
  float c   = lo + ((float)i * (hi - lo - 1.0f)) / 6.0f;
  float c0f = floorf(c);
  w  = c - c0f;
  i0 = (int)fminf(fmaxf(c0f,        0.0f), dm1);
  i1 = (int)fminf(fmaxf(c0f + 1.0f, 0.0f), dm1);
}

__global__ __launch_bounds__(BLOCK) void roi_align3d_kernel(
    const float* __restrict__ boxes,
    const float* __restrict__ fmap2,
    const float* __restrict__ fmap3,
    float* __restrict__ out,
    int C) {
  const int box = blockIdx.x;
  const int t   = threadIdx.x;
  if (t >= NPOS) return;

  // ---- per-box setup (scalarized to SALU-float by the compiler) ----
  const float b0 = boxes[box * 6 + 0];
  const float b1 = boxes[box * 6 + 1];
  const float b2 = boxes[box * 6 + 2];
  const float b3 = boxes[box * 6 + 3];
  const float b4 = boxes[box * 6 + 4];
  const float b5 = boxes[box * 6 + 5];

  const float dd = b3 - b0, hh = b4 - b1, ww = b5 - b2;
  float lvl = rintf(4.0f + log2f(hh * ww * dd) / 3.0f);  // RNE == jnp.round
  lvl = fminf(fmaxf(lvl, 2.0f), 3.0f);
  const bool  use2 = (lvl == 2.0f);

  const float* __restrict__ fmap = use2 ? fmap2 : fmap3;
  const int   Dd    = use2 ? 64 : 32;
  const float scale = (float)(Dd - 1);   // also the clamp bound (dim-1)

  const float zlo = floorf(b0 * scale);
  const float ylo = floorf(b1 * scale);
  const float xlo = floorf(b2 * scale);
  const float zhi = ceilf(b3 * scale + 1.0f);
  const float yhi = ceilf(b4 * scale + 1.0f);
  const float xhi = ceilf(b5 * scale + 1.0f);

  // thread -> (zi, yi, xi), xi fastest across lanes for row locality
  const int zi = t / 49;
  const int r  = t - zi * 49;
  const int yi = r / 7;
  const int xi = r - yi * 7;

  int z0, z1, y0, y1, x0, x1;
  float wz, wy, wx;
  axis_interp(zlo, zhi, zi, scale, z0, z1, wz);
  axis_interp(ylo, yhi, yi, scale, y0, y1, wy);
  axis_interp(xlo, xhi, xi, scale, x0, x1, wx);

  const int HW  = Dd * Dd;
  const int DHW = HW * Dd;
  const int b00 = z0 * HW + y0 * Dd;   // row bases for the 4 (z,y) corners
  const int b01 = z0 * HW + y1 * Dd;
  const int b10 = z1 * HW + y0 * Dd;
  const int b11 = z1 * HW + y1 * Dd;

  const float uz = 1.0f - wz, uy = 1.0f - wy, ux = 1.0f - wx;
  const float w000 = uz * uy * ux, w001 = uz * uy * wx;
  const float w010 = uz * wy * ux, w011 = uz * wy * wx;
  const float w100 = wz * uy * ux, w101 = wz * uy * wx;
  const float w110 = wz * wy * ux, w111 = wz * wy * wx;

  // ---- channel loop: 8 gathers + 8 FMA + 1 coalesced store per channel ----
  const int cbase = blockIdx.y * CPB;
  const float* __restrict__ fc = fmap + (size_t)cbase * (size_t)DHW;
  float* __restrict__ op =
      out + ((size_t)box * (size_t)C + (size_t)cbase) * (size_t)NPOS + t;

#pragma unroll 4
  for (int c = 0; c < CPB; ++c) {
    if (c + 1 < CPB) {
      // gfx1250 global_prefetch_b8: each line covers a corner row's x-span,
      // so 4 prefetches cover all 8 gathers of the next channel plane.
      __builtin_prefetch(fc + DHW + b00 + x0, 0, 3);
      __builtin_prefetch(fc + DHW + b01 + x0, 0, 3);
      __builtin_prefetch(fc + DHW + b10 + x0, 0, 3);
      __builtin_prefetch(fc + DHW + b11 + x0, 0, 3);
    }
    const float v000 = fc[b00 + x0];
    const float v001 = fc[b00 + x1];
    const float v010 = fc[b01 + x0];
    const float v011 = fc[b01 + x1];
    const float v100 = fc[b10 + x0];
    const float v101 = fc[b10 + x1];
    const float v110 = fc[b11 + x0];
    const float v111 = fc[b11 + x1];

    float acc = v000 * w000;
    acc = fmaf(v001, w001, acc);
    acc = fmaf(v010, w010, acc);
    acc = fmaf(v011, w011, acc);
    acc = fmaf(v100, w100, acc);
    acc = fmaf(v101, w101, acc);
    acc = fmaf(v110, w110, acc);
    acc = fmaf(v111, w111, acc);

    *op = acc;
    fc += DHW;
    op += NPOS;
  }
}

extern "C" void kernel_launch(void* const* d_in, const int* in_sizes, int n_in,
                              void* d_out, int out_size, void* d_ws, size_t ws_size,
                              hipStream_t stream) {
  const float* boxes = (const float*)d_in[0];
  const float* fmap2 = (const float*)d_in[1];
  const float* fmap3 = (const float*)d_in[2];
  float* out = (float*)d_out;

  const int n_boxes = in_sizes[0] / 6;              // 512
  const int C       = in_sizes[1] / (64 * 64 * 64); // 256
  dim3 grid(n_boxes, (C + CPB - 1) / CPB);          // (512, 8)

  roi_align3d_kernel<<<grid, BLOCK, 0, stream>>>(boxes, fmap2, fmap3, out, C);
}